// ExemplarBaseline_45603962749440
// MI455X (gfx1250) — compile-verified
//
#include <hip/hip_runtime.h>
#include <math.h>

typedef float v2f __attribute__((ext_vector_type(2)));
typedef float v8f __attribute__((ext_vector_type(8)));

#define EPS_NUMER 1e-08f
#define EPS_D2    1e-12f

// Problem constants (fixed by the reference setup)
#define BB   128      // batch
#define TT   128      // time
#define NCLS 10       // classes
#define DD   1024     // feature dim (GEMM N)
#define KK   3072     // C*H*W (GEMM K)
#define MM   (BB*TT)  // GEMM M = 16384

// ---------------------------------------------------------------------------
// Kernel 1: feats = X @ W + bias   (M=16384, K=3072, N=1024, fp32 WMMA)
// One BLOCK (8 waves) owns one 16-row m-tile across the FULL N=1024:
//   wave w computes the 16(M) x 128(N) strip at n_base = w*128 (8 accumulators).
// All 8 waves load the SAME A fragment -> X is pulled from HBM once and the
// other 7 waves hit WGP$/L0; W_fe (12.6 MB) lives in the 192 MB L2.
// HBM traffic ~= X (201 MB) + feats (67 MB) ~= 12 us at 23.3 TB/s.
//
// A layout (16x4 f32, ISA 7.12.2): lanes 0-15 -> M=lane, vgpr0=K0,vgpr1=K1;
//                                  lanes16-31 -> M=lane-16, vgpr0=K2,vgpr1=K3.
// B layout (4x16):  vgpr0 = row K0 (lanes0-15) / K2 (lanes16-31), vgpr1 = K1/K3.
// C/D layout (16x16 f32): N = lane&15, M = vgpr + 8*(lane>>4).
// ---------------------------------------------------------------------------
__global__ __launch_bounds__(256) void k_gemm_feats(
    const float* __restrict__ X, const float* __restrict__ W,
    const float* __restrict__ bias, float* __restrict__ feats)
{
  const int lane = threadIdx.x & 31;
  const int wave = threadIdx.x >> 5;
  const int m_tile = blockIdx.x;                // 1024 blocks, one m-tile each
  const int n_base = wave * 128;                // 8 waves cover N=1024
  const int ln = lane & 15;
  const int kh = lane >> 4;                     // 0 or 1 (K half)
  const int row = m_tile * 16 + ln;

  v8f acc[8] = {};
  const float* __restrict__ xrow = X + (size_t)row * KK;

  for (int k = 0; k < KK; k += 4) {
    // A fragment: contiguous pair X[row, k+2kh .. k+2kh+1] (same for all waves)
    v2f a = *(const v2f*)(xrow + k + 2 * kh);
    const int krow = k + 2 * kh;
    const float* __restrict__ w0 = W + (size_t)krow * DD + n_base + ln;
    const float* __restrict__ w1 = w0 + DD;     // next K row
    // keep the next A cacheline warm while WMMAs drain
    __builtin_prefetch(xrow + k + 32, 0, 1);
#pragma unroll
    for (int j = 0; j < 8; ++j) {
      v2f bf = { w0[j * 16], w1[j * 16] };
      acc[j] = __builtin_amdgcn_wmma_f32_16x16x4_f32(false, a, false, bf,
                                                     (short)0, acc[j], false, false);
    }
  }

  // Epilogue: D layout -> row = m_tile*16 + v + 8*kh, col = n_base + j*16 + ln
  const int rbase = m_tile * 16 + 8 * kh;
#pragma unroll
  for (int j = 0; j < 8; ++j) {
    const int col = n_base + j * 16 + ln;
    const float bv = bias[col];
#pragma unroll
    for (int v = 0; v < 8; ++v) {
      feats[(size_t)(rbase + v) * DD + col] = acc[j][v] + bv;
    }
  }
}

// ---------------------------------------------------------------------------
// Kernel 2: sq[row] = sum(feats[row,:]^2)   (16384 rows, one wave per row)
// ---------------------------------------------------------------------------
__global__ __launch_bounds__(256) void k_sqnorm(
    const float* __restrict__ feats, float* __restrict__ sq)
{
  const int lane = threadIdx.x & 31;
  const int row  = blockIdx.x * 8 + (threadIdx.x >> 5);
  const float4* __restrict__ fr = (const float4*)(feats + (size_t)row * DD);
  float s = 0.f;
#pragma unroll 4
  for (int i = lane; i < DD / 4; i += 32) {
    float4 v = fr[i];
    s += v.x * v.x + v.y * v.y + v.z * v.z + v.w * v.w;
  }
#pragma unroll
  for (int off = 16; off > 0; off >>= 1) s += __shfl_xor(s, off, 32);
  if (lane == 0) sq[row] = s;
}

// ---------------------------------------------------------------------------
// Kernel 3: per batch b, Gram G = f_b @ f_b^T (fp32 WMMA), fused into
// sims[b,s,t] = exp(-c * (max(sq_s + sq_t - 2G, eps))^(1/4)).
// One BLOCK per batch; wave w owns s-strip w*16 and computes all 8 t-tiles
// (A fragment loaded once per k-step, 8 WMMAs). Each batch's 512 KB of feats
// stays hot in WGP$/L2.
// ---------------------------------------------------------------------------
__global__ __launch_bounds__(256) void k_gram_sims(
    const float* __restrict__ feats, const float* __restrict__ sq,
    const float* __restrict__ c_ptr, float* __restrict__ sims)
{
  const int lane = threadIdx.x & 31;
  const int wave = threadIdx.x >> 5;
  const int b    = blockIdx.x;                  // 128 blocks
  const int s0   = wave * 16;                   // 8 waves cover T=128 rows
  const int ln = lane & 15;
  const int kh = lane >> 4;

  const float* __restrict__ f    = feats + (size_t)b * TT * DD;
  const float* __restrict__ arow = f + (size_t)(s0 + ln) * DD;
  const float* __restrict__ brow = f + (size_t)ln * DD;   // t-tile j adds j*16 rows

  v8f acc[8] = {};
  for (int k = 0; k < DD; k += 4) {
    v2f a = *(const v2f*)(arow + k + 2 * kh);   // A[M=s0+ln, K..K+1]
#pragma unroll
    for (int j = 0; j < 8; ++j) {
      // B[K..K+1, N=j*16+ln] = f[j*16+ln, k..k+1]
      v2f bf = *(const v2f*)(brow + (size_t)(j * 16) * DD + k + 2 * kh);
      acc[j] = __builtin_amdgcn_wmma_f32_16x16x4_f32(false, a, false, bf,
                                                     (short)0, acc[j], false, false);
    }
  }

  const float cc = c_ptr[0];
#pragma unroll
  for (int j = 0; j < 8; ++j) {
    const int t = j * 16 + ln;
    const float sqt = sq[b * TT + t];
#pragma unroll
    for (int v = 0; v < 8; ++v) {
      const int s = s0 + v + 8 * kh;
      const float d2 = fmaxf(sq[b * TT + s] + sqt - 2.f * acc[j][v], EPS_D2);
      const float dist = sqrtf(sqrtf(d2));
      sims[((size_t)b * TT + s) * TT + t] = expf(-cc * dist);
    }
  }
}

// ---------------------------------------------------------------------------
// Kernel 4: numers[b,t,c] = EPS + sum_{s<t} sims[b,s,t]*teach[b,s,c];
// tmp = numers^gamma; score = tmp / sum_c tmp; score[:,0,:] = EPS.
// One wave per (b,t).  NOTE: sims is symmetric (d2(s,t)=d2(t,s)), so we read
// ROW t (contiguous in s, fully coalesced) instead of column t.
// ---------------------------------------------------------------------------
__global__ __launch_bounds__(256) void k_score(
    const float* __restrict__ sims, const float* __restrict__ teach,
    const float* __restrict__ gamma_ptr, float* __restrict__ out)
{
  const int lane = threadIdx.x & 31;
  const int wave = threadIdx.x >> 5;
  const int job  = blockIdx.x * 8 + wave;       // B*T = 16384 jobs
  const int b = job / TT;
  const int t = job % TT;

  float acc[NCLS];
#pragma unroll
  for (int c = 0; c < NCLS; ++c) acc[c] = 0.f;

  // sims[b,t,s] == sims[b,s,t] by symmetry; row read is coalesced over lanes.
  const float* __restrict__ simrow = sims + ((size_t)b * TT + t) * TT;
  const float* __restrict__ tch    = teach + (size_t)b * TT * NCLS;

  for (int s = lane; s < t; s += 32) {
    const float sv = simrow[s];
    const float* __restrict__ tr = tch + s * NCLS;
#pragma unroll
    for (int c = 0; c < NCLS; ++c) acc[c] += sv * tr[c];
  }
#pragma unroll
  for (int off = 16; off > 0; off >>= 1) {
#pragma unroll
    for (int c = 0; c < NCLS; ++c) acc[c] += __shfl_xor(acc[c], off, 32);
  }

  if (lane == 0) {
    float* __restrict__ o = out + (size_t)job * NCLS;
    if (t == 0) {
#pragma unroll
      for (int c = 0; c < NCLS; ++c) o[c] = EPS_NUMER;
    } else {
      const float g = gamma_ptr[0];
      float tmp[NCLS];
      float sum = 0.f;
#pragma unroll
      for (int c = 0; c < NCLS; ++c) {
        const float p = powf(EPS_NUMER + acc[c], g);
        tmp[c] = p;
        sum += p;
      }
      const float inv = 1.f / sum;
#pragma unroll
      for (int c = 0; c < NCLS; ++c) o[c] = tmp[c] * inv;
    }
  }
}

// ---------------------------------------------------------------------------
// Launch
// Inputs (setup_inputs order): 0 responses_t (unused), 1 data_t, 2 teaching,
// 3 W_fe, 4 b_fe, 5 c, 6 gamma.  Output: (B,T,1,NC) float32.
// Workspace: feats | sq | sims (floats), ~76 MB total.
// ---------------------------------------------------------------------------
extern "C" void kernel_launch(void* const* d_in, const int* in_sizes, int n_in,
                              void* d_out, int out_size, void* d_ws, size_t ws_size,
                              hipStream_t stream) {
  (void)in_sizes; (void)n_in; (void)out_size; (void)ws_size;
  const float* X     = (const float*)d_in[1];   // data_t  (B*T, 3072)
  const float* teach = (const float*)d_in[2];   // (B, T, NC)
  const float* Wfe   = (const float*)d_in[3];   // (3072, 1024)
  const float* bfe   = (const float*)d_in[4];   // (1024,)
  const float* cC    = (const float*)d_in[5];
  const float* gamma = (const float*)d_in[6];
  float* out = (float*)d_out;

  float* ws    = (float*)d_ws;
  float* feats = ws;                                   // MM*DD   = 16,777,216 floats
  float* sq    = feats + (size_t)MM * DD;              // MM      = 16,384 floats
  float* sims  = sq + MM;                              // BB*TT*TT= 2,097,152 floats

  // One block per m-tile; 8 waves cover full N
  k_gemm_feats<<<1024, 256, 0, stream>>>(X, Wfe, bfe, feats);
  // 16384 rows / 8 per block
  k_sqnorm<<<2048, 256, 0, stream>>>(feats, sq);
  // One block per batch; 8 waves cover all s-strips
  k_gram_sims<<<128, 256, 0, stream>>>(feats, sq, cC, sims);
  // 16384 wave-jobs / 8 per block
  k_score<<<2048, 256, 0, stream>>>(sims, teach, gamma, out);
}